// GaussianRegistration_3702261809882
// MI455X (gfx1250) — compile-verified
//
#include <hip/hip_runtime.h>
#include <math.h>

#define INV_EPS 10.0f      // 1/SINK_EPS
#define MAX_ITER 100
#define LAMBDA_REG 0.1f
#define FDIM 100           // padded feature stride: 4 (mean) + 12 (cov) + 84 (cov x cov)

typedef float v2f __attribute__((ext_vector_type(2)));
typedef float v8f __attribute__((ext_vector_type(8)));

// ---------------------------------------------------------------------------
// Setup: se3_exp(xi) -> R,t ; weight sums, log weights, reg sum, lu/lv init.
// scal layout: [0..8]=R row-major, [9..11]=t, [12]=sumA, [13]=sumB,
//              [14]=regsum, [15]=global max (written later)
// ---------------------------------------------------------------------------
__global__ void k_setup(const float* __restrict__ xi,
                        const float* __restrict__ src_w, const float* __restrict__ tgt_w,
                        const int* __restrict__ idxS, const int* __restrict__ idxT, int B,
                        float* __restrict__ scal,
                        float* __restrict__ log_a, float* __restrict__ log_b,
                        float* __restrict__ lu, float* __restrict__ lv) {
  __shared__ float red[256];
  __shared__ float s_sumA, s_sumB;
  int tid = threadIdx.x;

  if (tid == 0) {
    float wx = xi[0], wy = xi[1], wz = xi[2];
    float t0 = xi[3], t1 = xi[4], t2 = xi[5];
    float ang = sqrtf(wx * wx + wy * wy + wz * wz);
    float R[9], tt[3];
    if (ang < 1e-6f) {
      R[0] = 1.f; R[1] = 0.f; R[2] = 0.f;
      R[3] = 0.f; R[4] = 1.f; R[5] = 0.f;
      R[6] = 0.f; R[7] = 0.f; R[8] = 1.f;
      tt[0] = t0; tt[1] = t1; tt[2] = t2;
    } else {
      float ax = wx / ang, ay = wy / ang, az = wz / ang;
      float s = sinf(ang), c = cosf(ang), C1 = 1.0f - c;
      R[0] = c + ax * ax * C1;      R[1] = ax * ay * C1 - az * s; R[2] = ax * az * C1 + ay * s;
      R[3] = ay * ax * C1 + az * s; R[4] = c + ay * ay * C1;      R[5] = ay * az * C1 - ax * s;
      R[6] = az * ax * C1 - ay * s; R[7] = az * ay * C1 + ax * s; R[8] = c + az * az * C1;
      float sa = s / ang, oa = 1.0f - sa, ka = C1 / ang;
      float J[9];
      J[0] = sa + oa * ax * ax;     J[1] = oa * ax * ay - ka * az; J[2] = oa * ax * az + ka * ay;
      J[3] = oa * ay * ax + ka * az; J[4] = sa + oa * ay * ay;     J[5] = oa * ay * az - ka * ax;
      J[6] = oa * az * ax - ka * ay; J[7] = oa * az * ay + ka * ax; J[8] = sa + oa * az * az;
      tt[0] = J[0] * t0 + J[1] * t1 + J[2] * t2;
      tt[1] = J[3] * t0 + J[4] * t1 + J[5] * t2;
      tt[2] = J[6] * t0 + J[7] * t1 + J[8] * t2;
    }
    for (int k = 0; k < 9; ++k) scal[k] = R[k];
    scal[9] = tt[0]; scal[10] = tt[1]; scal[11] = tt[2];
  }

  // sum of selected source weights
  float pa = 0.f, pb = 0.f;
  for (int i = tid; i < B; i += 256) {
    pa += src_w[idxS[i]];
    pb += tgt_w[idxT[i]];
  }
  red[tid] = pa; __syncthreads();
  for (int o = 128; o > 0; o >>= 1) { if (tid < o) red[tid] += red[tid + o]; __syncthreads(); }
  if (tid == 0) { s_sumA = red[0]; scal[12] = red[0]; }
  __syncthreads();
  red[tid] = pb; __syncthreads();
  for (int o = 128; o > 0; o >>= 1) { if (tid < o) red[tid] += red[tid + o]; __syncthreads(); }
  if (tid == 0) { s_sumB = red[0]; scal[13] = red[0]; }
  __syncthreads();

  float sumA = s_sumA, sumB = s_sumB;
  float preg = 0.f;
  for (int i = tid; i < B; i += 256) {
    float wa = src_w[idxS[i]] / sumA;
    float wb = tgt_w[idxT[i]] / sumB;
    log_a[i] = logf(wa + 1e-10f);
    log_b[i] = logf(wb + 1e-10f);
    lu[i] = 0.f; lv[i] = 0.f;
    preg += fabsf(wa);
  }
  red[tid] = preg; __syncthreads();
  for (int o = 128; o > 0; o >>= 1) { if (tid < o) red[tid] += red[tid + o]; __syncthreads(); }
  if (tid == 0) scal[14] = red[0];
}

// ---------------------------------------------------------------------------
// Per-point features. Row layout (FDIM=100):
//  [0..2] mean (3) [3]=0
//  [4..12] cov row-major (9), [13..15]=0
//  [16..96] outer features (81): side A: f_k = A[p][r]*A[q][s],
//                                side B: g_k = B[r][q]*B[s][p],
//           k = ((p*3+r)*3+q)*3+s ; [97..99]=0
// dot(fA,gB) over segs = (a.b), tr(AB), tr(ABAB).
// ---------------------------------------------------------------------------
__global__ void k_features(const float* __restrict__ mu, const float* __restrict__ cov,
                           const int* __restrict__ idx, int B,
                           const float* __restrict__ scal, int side,
                           float* __restrict__ F, float* __restrict__ nrm,
                           float* __restrict__ trc, float* __restrict__ det) {
  int i = blockIdx.x * blockDim.x + threadIdx.x;
  if (i >= B) return;
  int j = idx[i];
  float m0 = mu[3 * j + 0], m1 = mu[3 * j + 1], m2 = mu[3 * j + 2];
  float Cin[3][3];
#pragma unroll
  for (int p = 0; p < 3; ++p)
#pragma unroll
    for (int r = 0; r < 3; ++r) Cin[p][r] = cov[9 * j + 3 * p + r];

  float a0, a1, a2;
  float A[3][3];
  if (side == 0) {
    float R[3][3], t[3];
#pragma unroll
    for (int p = 0; p < 3; ++p) {
#pragma unroll
      for (int r = 0; r < 3; ++r) R[p][r] = scal[3 * p + r];
      t[p] = scal[9 + p];
    }
    a0 = R[0][0] * m0 + R[0][1] * m1 + R[0][2] * m2 + t[0];
    a1 = R[1][0] * m0 + R[1][1] * m1 + R[1][2] * m2 + t[1];
    a2 = R[2][0] * m0 + R[2][1] * m1 + R[2][2] * m2 + t[2];
    float T[3][3];
#pragma unroll
    for (int p = 0; p < 3; ++p)
#pragma unroll
      for (int r = 0; r < 3; ++r)
        T[p][r] = R[p][0] * Cin[0][r] + R[p][1] * Cin[1][r] + R[p][2] * Cin[2][r];
#pragma unroll
    for (int p = 0; p < 3; ++p)
#pragma unroll
      for (int r = 0; r < 3; ++r)
        A[p][r] = T[p][0] * R[r][0] + T[p][1] * R[r][1] + T[p][2] * R[r][2];
  } else {
    a0 = m0; a1 = m1; a2 = m2;
#pragma unroll
    for (int p = 0; p < 3; ++p)
#pragma unroll
      for (int r = 0; r < 3; ++r) A[p][r] = Cin[p][r];
  }

  float* f = F + (size_t)i * FDIM;
  f[0] = a0; f[1] = a1; f[2] = a2; f[3] = 0.f;
#pragma unroll
  for (int p = 0; p < 3; ++p)
#pragma unroll
    for (int r = 0; r < 3; ++r) f[4 + 3 * p + r] = A[p][r];
  f[13] = 0.f; f[14] = 0.f; f[15] = 0.f;
#pragma unroll
  for (int p = 0; p < 3; ++p)
#pragma unroll
    for (int r = 0; r < 3; ++r)
#pragma unroll
      for (int q = 0; q < 3; ++q)
#pragma unroll
        for (int s = 0; s < 3; ++s) {
          int k = ((p * 3 + r) * 3 + q) * 3 + s;
          float val = (side == 0) ? (A[p][r] * A[q][s]) : (A[r][q] * A[s][p]);
          f[16 + k] = val;
        }
  f[97] = 0.f; f[98] = 0.f; f[99] = 0.f;

  nrm[i] = a0 * a0 + a1 * a1 + a2 * a2;
  trc[i] = A[0][0] + A[1][1] + A[2][2];
  det[i] = A[0][0] * (A[1][1] * A[2][2] - A[1][2] * A[2][1])
         - A[0][1] * (A[1][0] * A[2][2] - A[1][2] * A[2][0])
         + A[0][2] * (A[1][0] * A[2][1] - A[1][1] * A[2][0]);
}

// ---------------------------------------------------------------------------
// WMMA cost-matrix kernel: one wave (32 lanes) per 16x16 output tile.
// 25x v_wmma_f32_16x16x4_f32 per tile, three accumulators (mean dot, tr(AB),
// tr(ABAB)). Epilogue solves the eigen-cubic of A*B analytically.
// A-frag and B-frag for 16x16x4 f32: lane%16 = row index, VGPR pair holds
// K = k0+2*(lane/16), k0+2*(lane/16)+1 -> one 8-byte load per fragment.
// ---------------------------------------------------------------------------
__global__ void k_cost(const float* __restrict__ FA, const float* __restrict__ FB,
                       const float* __restrict__ naA, const float* __restrict__ trA,
                       const float* __restrict__ detA,
                       const float* __restrict__ naB, const float* __restrict__ trB,
                       const float* __restrict__ detB,
                       int B, float* __restrict__ Cm, float* __restrict__ Ct) {
  int ntj = B >> 4;
  int ti = blockIdx.x / ntj;
  int tj = blockIdx.x - ti * ntj;
  int lane = threadIdx.x;
  int half = lane >> 4;
  int lm = lane & 15;

  const float* pa = FA + (size_t)(ti * 16 + lm) * FDIM + 2 * half;
  const float* pb = FB + (size_t)(tj * 16 + lm) * FDIM + 2 * half;

  v8f cM = {};
  {
    v2f av = *(const v2f*)(pa + 0);
    v2f bv = *(const v2f*)(pb + 0);
    cM = __builtin_amdgcn_wmma_f32_16x16x4_f32(false, av, false, bv, (short)0, cM, false, false);
  }
  v8f c1 = {};
#pragma unroll
  for (int k0 = 4; k0 < 16; k0 += 4) {
    v2f av = *(const v2f*)(pa + k0);
    v2f bv = *(const v2f*)(pb + k0);
    c1 = __builtin_amdgcn_wmma_f32_16x16x4_f32(false, av, false, bv, (short)0, c1, false, false);
  }
  v8f c2 = {};
#pragma unroll
  for (int k0 = 16; k0 < 100; k0 += 4) {
    v2f av = *(const v2f*)(pa + k0);
    v2f bv = *(const v2f*)(pb + k0);
    c2 = __builtin_amdgcn_wmma_f32_16x16x4_f32(false, av, false, bv, (short)0, c2, false, false);
  }

  int j = tj * 16 + lm;
  float nb = naB[j], tb = trB[j], db = detB[j];
  const float TWO_PI_3 = 2.0943951023931953f;

#pragma unroll
  for (int v = 0; v < 8; ++v) {
    int m = v + 8 * half;
    int i = ti * 16 + m;
    float na = naA[i], ta = trA[i], da = detA[i];
    float mdot = cM[v];
    float s1 = c1[v];        // tr(A*B)
    float s2 = c2[v];        // tr(A*B*A*B)
    float I1 = s1;
    float I2 = 0.5f * (s1 * s1 - s2);
    float I3 = da * db;
    // depressed cubic x^3 + p x + q = 0, lambda = x + I1/3
    float p = I2 - I1 * I1 * (1.0f / 3.0f);
    float q = I1 * I2 * (1.0f / 3.0f) - I3 - (2.0f / 27.0f) * (I1 * I1 * I1);
    float r = sqrtf(fmaxf(-p, 0.0f) * (1.0f / 3.0f));
    float rr = fmaxf(r, 1e-20f);
    float arg = -q / (2.0f * rr * rr * rr);
    arg = fminf(fmaxf(arg, -1.0f), 1.0f);
    float th = acosf(arg) * (1.0f / 3.0f);
    float lc = I1 * (1.0f / 3.0f);
    float x0 = 2.0f * r * cosf(th);
    float x1 = 2.0f * r * cosf(th - TWO_PI_3);
    float x2 = 2.0f * r * cosf(th + TWO_PI_3);
    float trs = sqrtf(fmaxf(lc + x0, 0.0f))
              + sqrtf(fmaxf(lc + x1, 0.0f))
              + sqrtf(fmaxf(lc + x2, 0.0f));
    float mean_t = na + nb - 2.0f * mdot;
    float cov_t  = ta + tb - 2.0f * trs;
    float cval = mean_t + cov_t;
    Cm[(size_t)i * B + j] = cval;
    Ct[(size_t)j * B + i] = cval;
  }
}

// ---------------------------------------------------------------------------
// One Sinkhorn half-step: out[r] = logw[r] - LSE_j( vin[j] - Cmat[r,j]/eps )
// One 256-thread block per row, online (max,sum) LSE + LDS tree reduce.
// ---------------------------------------------------------------------------
__global__ void k_lse_pass(const float* __restrict__ Cmat, const float* __restrict__ vin,
                           const float* __restrict__ logw, float* __restrict__ out, int B) {
  int r = blockIdx.x;
  int tid = threadIdx.x;
  const float* row = Cmat + (size_t)r * B;
  float m = -__builtin_inff();
  float s = 0.0f;
  for (int j = tid; j < B; j += 256) {
    float x = vin[j] - row[j] * INV_EPS;
    if (x > m) { s = s * __expf(m - x) + 1.0f; m = x; }
    else s += __expf(x - m);
  }
  __shared__ float sm[256], ss[256];
  sm[tid] = m; ss[tid] = s;
  __syncthreads();
  for (int o = 128; o > 0; o >>= 1) {
    if (tid < o) {
      float m1 = sm[tid], m2 = sm[tid + o];
      float M = fmaxf(m1, m2);
      float sn;
      if (M == -__builtin_inff()) sn = 0.0f;
      else sn = ss[tid] * __expf(m1 - M) + ss[tid + o] * __expf(m2 - M);
      sm[tid] = M; ss[tid] = sn;
    }
    __syncthreads();
  }
  if (tid == 0) out[r] = logw[r] - (sm[0] + logf(ss[0]));
}

// ---------------------------------------------------------------------------
// Final loss: M = max(lu_i + lv_j - C_ij/eps); loss = sum exp(..-M)*C + reg
// ---------------------------------------------------------------------------
__global__ void k_partial_max(const float* __restrict__ Cm, const float* __restrict__ lu,
                              const float* __restrict__ lv, int B, float* __restrict__ part) {
  int tid = threadIdx.x;
  int gid = blockIdx.x * 256 + tid;
  int stride = gridDim.x * 256;
  int total = B * B;
  float m = -__builtin_inff();
  for (int e = gid; e < total; e += stride) {
    int i = e / B;
    int j = e - i * B;
    float x = lu[i] + lv[j] - Cm[e] * INV_EPS;
    m = fmaxf(m, x);
  }
  __shared__ float sm[256];
  sm[tid] = m; __syncthreads();
  for (int o = 128; o > 0; o >>= 1) { if (tid < o) sm[tid] = fmaxf(sm[tid], sm[tid + o]); __syncthreads(); }
  if (tid == 0) part[blockIdx.x] = sm[0];
}

__global__ void k_final_max(const float* __restrict__ part, int n, float* __restrict__ scal) {
  int tid = threadIdx.x;
  __shared__ float sm[256];
  float m = -__builtin_inff();
  for (int i = tid; i < n; i += 256) m = fmaxf(m, part[i]);
  sm[tid] = m; __syncthreads();
  for (int o = 128; o > 0; o >>= 1) { if (tid < o) sm[tid] = fmaxf(sm[tid], sm[tid + o]); __syncthreads(); }
  if (tid == 0) scal[15] = sm[0];
}

__global__ void k_partial_sum(const float* __restrict__ Cm, const float* __restrict__ lu,
                              const float* __restrict__ lv, const float* __restrict__ scal,
                              int B, float* __restrict__ part) {
  int tid = threadIdx.x;
  int gid = blockIdx.x * 256 + tid;
  int stride = gridDim.x * 256;
  int total = B * B;
  float g = scal[15];
  float s = 0.0f;
  for (int e = gid; e < total; e += stride) {
    int i = e / B;
    int j = e - i * B;
    float c = Cm[e];
    float x = lu[i] + lv[j] - c * INV_EPS - g;
    s += __expf(x) * c;
  }
  __shared__ float sm[256];
  sm[tid] = s; __syncthreads();
  for (int o = 128; o > 0; o >>= 1) { if (tid < o) sm[tid] += sm[tid + o]; __syncthreads(); }
  if (tid == 0) part[blockIdx.x] = sm[0];
}

__global__ void k_final_sum(const float* __restrict__ part, int n,
                            const float* __restrict__ scal, float* __restrict__ out) {
  int tid = threadIdx.x;
  __shared__ float sm[256];
  float s = 0.0f;
  for (int i = tid; i < n; i += 256) s += part[i];
  sm[tid] = s; __syncthreads();
  for (int o = 128; o > 0; o >>= 1) { if (tid < o) sm[tid] += sm[tid + o]; __syncthreads(); }
  if (tid == 0) out[0] = sm[0] + LAMBDA_REG * scal[14];
}

// ---------------------------------------------------------------------------
extern "C" void kernel_launch(void* const* d_in, const int* in_sizes, int n_in,
                              void* d_out, int out_size, void* d_ws, size_t ws_size,
                              hipStream_t stream) {
  const float* xi   = (const float*)d_in[0];
  const float* smu  = (const float*)d_in[1];
  const float* scov = (const float*)d_in[2];
  const float* sw   = (const float*)d_in[3];
  const float* tmu  = (const float*)d_in[4];
  const float* tcov = (const float*)d_in[5];
  const float* tw   = (const float*)d_in[6];
  const int* idxS   = (const int*)d_in[7];
  const int* idxT   = (const int*)d_in[8];
  int B = in_sizes[7];            // 2000 (multiple of 16)

  float* ws = (float*)d_ws;
  float* scal  = ws;              // 64 floats reserved
  float* log_a = ws + 64;
  float* log_b = log_a + B;
  float* lu    = log_b + B;
  float* lv    = lu + B;
  float* naA   = lv + B;
  float* trA   = naA + B;
  float* detA  = trA + B;
  float* naB   = detA + B;
  float* trB   = naB + B;
  float* detB  = trB + B;
  float* part  = detB + B;        // 4096 floats
  size_t off = (size_t)64 + (size_t)10 * B + 4096;
  off = (off + 15) & ~(size_t)15;
  float* FA = ws + off;
  float* FB = FA + (size_t)B * FDIM;
  float* Cm = FB + (size_t)B * FDIM;
  float* Ct = Cm + (size_t)B * B;

  k_setup<<<1, 256, 0, stream>>>(xi, sw, tw, idxS, idxT, B, scal, log_a, log_b, lu, lv);

  int fb = (B + 255) / 256;
  k_features<<<fb, 256, 0, stream>>>(smu, scov, idxS, B, scal, 0, FA, naA, trA, detA);
  k_features<<<fb, 256, 0, stream>>>(tmu, tcov, idxT, B, scal, 1, FB, naB, trB, detB);

  int nt = B / 16;
  k_cost<<<nt * nt, 32, 0, stream>>>(FA, FB, naA, trA, detA, naB, trB, detB, B, Cm, Ct);

  for (int it = 0; it < MAX_ITER; ++it) {
    k_lse_pass<<<B, 256, 0, stream>>>(Cm, lv, log_a, lu, B);
    k_lse_pass<<<B, 256, 0, stream>>>(Ct, lu, log_b, lv, B);
  }

  k_partial_max<<<256, 256, 0, stream>>>(Cm, lu, lv, B, part);
  k_final_max<<<1, 256, 0, stream>>>(part, 256, scal);
  k_partial_sum<<<256, 256, 0, stream>>>(Cm, lu, lv, scal, B, part);
  k_final_sum<<<1, 256, 0, stream>>>(part, 256, scal, (float*)d_out);
}